// HeatmapLayer_14104672600312
// MI455X (gfx1250) — compile-verified
//
#include <hip/hip_runtime.h>
#include <math.h>

// HeatmapLayer: out[b,c,i,j] = exp(-0.5*(dx2'[i] + dy2'[j])) = ex[i]*ey[j]
// -> rank-1 outer product per (b,c), via V_WMMA_F32_16X16X4_F32
//    (K=4 zero-padded; only K=0 carries data).
// Store-bandwidth bound: 56.6 MB fp32 out @ 23.3 TB/s ~ 2.4 us floor.

typedef __attribute__((ext_vector_type(2))) float v2f;
typedef __attribute__((ext_vector_type(8))) float v8f;

#define B_  16
#define C_  6
#define H_  384
#define W_  384

__global__ __launch_bounds__(128)
void heatmap_wmma_kernel(const float* __restrict__ labels,
                         const float* __restrict__ log_weight,
                         float* __restrict__ out) {
    __shared__ float s_ex[16];
    __shared__ float s_ey[W_ + 16];   // last 16 floats = zero pad for lanes 16-31

    const int blk   = blockIdx.x;
    const int itile = blk % (H_ / 16);
    const int bc    = blk / (H_ / 16);
    const int b     = bc / C_;
    const int c     = bc % C_;
    const int i0    = itile * 16;

    // Per-(b,c) scalars (uniform across block).
    const float mx    = labels[b * 12 + 2 * c]     * (float)H_;
    const float my    = labels[b * 12 + 2 * c + 1] * (float)W_;
    const float sigma = (float)H_ * expf(log_weight[0]);
    const float s20   = 20.0f * sigma;

    // Row minima: (i-m)^2 convex in i -> integer argmin at floor(m) or
    // floor(m)+1, clamped to grid. Same fp32 ops (sub, mul, div) as reference.
    int ifx = (int)floorf(mx);
    int cx0 = ifx < 0 ? 0 : (ifx > H_ - 1 ? H_ - 1 : ifx);
    int cx1 = (ifx + 1) < 0 ? 0 : ((ifx + 1) > H_ - 1 ? H_ - 1 : (ifx + 1));
    float dxa = (float)cx0 - mx, dxb = (float)cx1 - mx;
    const float dx2min = fminf(dxa * dxa / sigma, dxb * dxb / sigma);

    int ify = (int)floorf(my);
    int cy0 = ify < 0 ? 0 : (ify > W_ - 1 ? W_ - 1 : ify);
    int cy1 = (ify + 1) < 0 ? 0 : ((ify + 1) > W_ - 1 ? W_ - 1 : (ify + 1));
    float dya = (float)cy0 - my, dyb = (float)cy1 - my;
    const float dy2min = fminf(dya * dya / s20, dyb * dyb / s20);

    // Precompute separable exp factors into LDS.
    const int t = threadIdx.x;
    if (t < 16) {
        float d = (float)(i0 + t) - mx;
        s_ex[t]       = expf(-0.5f * (d * d / sigma - dx2min));
        s_ey[W_ + t]  = 0.0f;         // zero pad
    }
    for (int j = t; j < W_; j += 128) {
        float d = (float)j - my;
        s_ey[j] = expf(-0.5f * (d * d / s20 - dy2min));
    }
    __syncthreads();

    const int wave = t >> 5;
    const int lane = t & 31;
    const int col  = lane & 15;
    const bool lo  = (lane < 16);

    // A-matrix (16x4 f32): lanes 0-15, VGPR0 = column K=0 = ex tile; rest 0.
    v2f a;
    a.x = lo ? s_ex[col] : 0.0f;
    a.y = 0.0f;

    // B-matrix feed: lanes 0-15 walk the ey tiles (stride 64 floats = 4 tiles
    // round-robin over 4 waves); lanes 16-31 pin to the zero pad (stride 0).
    // -> inner loop has no selects / EXEC games: pure ds_load + wmma + stores.
    const float* pb   = lo ? (s_ey + wave * 16 + col) : (s_ey + W_ + col);
    const int bstride = lo ? 64 : 0;

    float* base = out + ((size_t)bc * H_ + i0) * (size_t)W_;
    const int row_off = lo ? 0 : 8;
    float* p = base + wave * 16 + col;

#pragma unroll
    for (int k = 0; k < (W_ / 16) / 4; ++k) {   // 6 tiles per wave
        v2f bm;
        bm.x = *pb;
        bm.y = 0.0f;

        v8f cm = {0.0f, 0.0f, 0.0f, 0.0f, 0.0f, 0.0f, 0.0f, 0.0f};
        // (neg_a, A, neg_b, B, c_mod, C, reuse_a, reuse_b)
        v8f d = __builtin_amdgcn_wmma_f32_16x16x4_f32(
            false, a, false, bm, (short)0, cm, false, false);

        // D layout (16x16 f32): lane L, VGPR r -> row = r + 8*(L>=16), col = L&15.
#pragma unroll
        for (int r = 0; r < 8; ++r) {
            p[(size_t)(r + row_off) * W_] = d[r];
        }

        pb += bstride;   // next tile for this wave (lanes 16-31 stay on zeros)
        p  += 64;        // 4 waves * 16 cols
    }
}

extern "C" void kernel_launch(void* const* d_in, const int* in_sizes, int n_in,
                              void* d_out, int out_size, void* d_ws, size_t ws_size,
                              hipStream_t stream) {
    // d_in[0] = x (only shape matters), d_in[1] = labels (16x12 f32),
    // d_in[2] = log_weight (1 f32). Output: (16, 6, 384, 384) f32.
    const float* labels     = (const float*)d_in[1];
    const float* log_weight = (const float*)d_in[2];
    float* out              = (float*)d_out;

    dim3 grid(B_ * C_ * (H_ / 16));  // 2304 blocks
    heatmap_wmma_kernel<<<grid, 128, 0, stream>>>(labels, log_weight, out);
}